// Decoder_4982162063741
// MI455X (gfx1250) — compile-verified
//
#include <hip/hip_runtime.h>

// ---------------------------------------------------------------------------
// Pointer-network decode step, algebraically folded + batched WMMA GEMMs.
//   B=2048, N=100, DIM=512, H=8, dh=64
//
// prep:   Wk_comb = Wk_mha@W_K1 ; WvT = (Wv_mha@W_V)^T ; M1 = W_Q@Wo_mha
//         A1T=(Wq@W_target)^T ; A2T=(Wq@W_global)^T ; Ffold = M1^T@W_K2
// batched: TGT gather -> Q = TGT@A1^T + G@A2^T -> S_h = Q_h @ Wk_comb_h
//          attn kernel: scores=0.125*S@E^T, softmax, C_h = attn_h@E
//          MHA_h = C_h @ WvT[:,h-block] -> QK2 = MHA@Ffold * 1/sqrt(512)
//          logits[n] = 10*tanh(E_n.qk2) - mask*1e9 ; log_softmax
// ---------------------------------------------------------------------------

typedef float v2f __attribute__((ext_vector_type(2)));
typedef float v8f __attribute__((ext_vector_type(8)));

#define DIM    512
#define NNODE  100
#define BATCH  2048

__device__ inline float wave_reduce_sum(float v) {
    for (int o = 16; o > 0; o >>= 1) v += __shfl_xor(v, o, 32);
    return v;
}
__device__ inline float wave_reduce_max(float v) {
    for (int o = 16; o > 0; o >>= 1) v = fmaxf(v, __shfl_xor(v, o, 32));
    return v;
}

// ---------------- prep: 16x16 tile of C = op(A)@B, K=512 -------------------
// A frag (16x4 MxK): lane L<16 -> M=L, K={0,1}; lanes>=16 -> K={2,3}.
// B frag (4x16 KxN): lane L -> N=L%16, K=2*(L>=16)+{0,1}.
// D: vgpr r, lanes 0-15 -> M=r,N=lane ; lanes 16-31 -> M=r+8,N=lane-16.
__device__ inline void gemm_tile_f32(const float* __restrict__ A,
                                     const float* __restrict__ B,
                                     float* __restrict__ Dst,
                                     int i0, int j0, int transA, int transD,
                                     int lane)
{
    const int hi = lane >> 4;
    const int lm = lane & 15;
    v8f c = {};
    for (int t0 = 0; t0 < DIM; t0 += 4) {
        const int k0 = t0 + hi * 2;
        v2f a, b;
        if (transA) {
            a.x = A[(k0    ) * DIM + i0 + lm];
            a.y = A[(k0 + 1) * DIM + i0 + lm];
        } else {
            a.x = A[(i0 + lm) * DIM + k0    ];
            a.y = A[(i0 + lm) * DIM + k0 + 1];
        }
        b.x = B[(k0    ) * DIM + j0 + lm];
        b.y = B[(k0 + 1) * DIM + j0 + lm];
        c = __builtin_amdgcn_wmma_f32_16x16x4_f32(false, a, false, b,
                                                  (short)0, c, false, false);
    }
    for (int r = 0; r < 8; ++r) {
        const int m = r + hi * 8;
        if (transD) Dst[(j0 + lm) * DIM + i0 + m] = c[r];
        else        Dst[(i0 + m) * DIM + j0 + lm] = c[r];
    }
}

__global__ __launch_bounds__(256)
void prep1(const float* __restrict__ Wt,  const float* __restrict__ Wg,
           const float* __restrict__ Wk1, const float* __restrict__ Wq,
           const float* __restrict__ Wv,  const float* __restrict__ Wqm,
           const float* __restrict__ Wkm, const float* __restrict__ Wvm,
           const float* __restrict__ Wom,
           float* __restrict__ wkcomb, float* __restrict__ wvT,
           float* __restrict__ m1,     float* __restrict__ a1T,
           float* __restrict__ a2T)
{
    const int wid  = threadIdx.x >> 5;
    const int lane = threadIdx.x & 31;
    const int gid  = blockIdx.x * 8 + wid;
    const int g    = gid >> 10;
    const int tile = gid & 1023;
    const int i0   = (tile >> 5) * 16;
    const int j0   = (tile & 31) * 16;

    const float *A, *B;
    float* Dp;
    int tD = 0;
    switch (g) {
        case 0:  A = Wkm; B = Wk1; Dp = wkcomb;         break;
        case 1:  A = Wvm; B = Wv;  Dp = wvT;    tD = 1; break;
        case 2:  A = Wq;  B = Wom; Dp = m1;             break;
        case 3:  A = Wqm; B = Wt;  Dp = a1T;    tD = 1; break;
        default: A = Wqm; B = Wg;  Dp = a2T;    tD = 1; break;
    }
    gemm_tile_f32(A, B, Dp, i0, j0, 0, tD, lane);
}

__global__ __launch_bounds__(256)
void prep2(const float* __restrict__ m1, const float* __restrict__ Wk2,
           float* __restrict__ ffold)
{
    const int wid  = threadIdx.x >> 5;
    const int lane = threadIdx.x & 31;
    const int gid  = blockIdx.x * 8 + wid;
    const int i0   = ((gid >> 5) & 31) * 16;
    const int j0   = (gid & 31) * 16;
    gemm_tile_f32(m1, Wk2, ffold, i0, j0, /*transA=*/1, /*transD=*/0, lane);
}

// ---------------- batched WMMA GEMM: D = A@B (+ A2@B2), 16x64 per wave -----
// B is stored [k][n]. blockIdx.y selects a head block via aStep/bStep/dStep.
__global__ __launch_bounds__(256)
void gemm_wmma(const float* __restrict__ A,  int lda, int aStep,
               const float* __restrict__ B,  int ldb, int bStep,
               float* __restrict__ Dst,      int ldd, int dStep,
               int nColT, int K,
               const float* __restrict__ A2, const float* __restrict__ B2,
               float scale)
{
    const int wid  = threadIdx.x >> 5;
    const int lane = threadIdx.x & 31;
    const int hi   = lane >> 4;
    const int lm   = lane & 15;
    const int job  = blockIdx.x * 8 + wid;
    const int i0   = (job / nColT) * 16;
    const int j0   = (job % nColT) * 64;

    A   += (size_t)blockIdx.y * aStep;
    B   += (size_t)blockIdx.y * bStep;
    Dst += (size_t)blockIdx.y * dStep;

    v8f c[4];
#pragma unroll
    for (int q = 0; q < 4; ++q) c[q] = (v8f){};

    for (int t0 = 0; t0 < K; t0 += 4) {
        const int k0 = t0 + hi * 2;
        v2f a;
        a.x = A[(size_t)(i0 + lm) * lda + k0];
        a.y = A[(size_t)(i0 + lm) * lda + k0 + 1];
#pragma unroll
        for (int q = 0; q < 4; ++q) {
            v2f b;
            b.x = B[(size_t)(k0    ) * ldb + j0 + q * 16 + lm];
            b.y = B[(size_t)(k0 + 1) * ldb + j0 + q * 16 + lm];
            c[q] = __builtin_amdgcn_wmma_f32_16x16x4_f32(false, a, false, b,
                                                         (short)0, c[q],
                                                         false, false);
        }
    }
    if (A2 != nullptr) {
        for (int t0 = 0; t0 < K; t0 += 4) {
            const int k0 = t0 + hi * 2;
            v2f a;
            a.x = A2[(size_t)(i0 + lm) * lda + k0];
            a.y = A2[(size_t)(i0 + lm) * lda + k0 + 1];
#pragma unroll
            for (int q = 0; q < 4; ++q) {
                v2f b;
                b.x = B2[(size_t)(k0    ) * ldb + j0 + q * 16 + lm];
                b.y = B2[(size_t)(k0 + 1) * ldb + j0 + q * 16 + lm];
                c[q] = __builtin_amdgcn_wmma_f32_16x16x4_f32(false, a, false, b,
                                                             (short)0, c[q],
                                                             false, false);
            }
        }
    }
#pragma unroll
    for (int r = 0; r < 8; ++r) {
        const int m = r + hi * 8;
#pragma unroll
        for (int q = 0; q < 4; ++q)
            Dst[(size_t)(i0 + m) * ldd + j0 + q * 16 + lm] = c[q][r] * scale;
    }
}

// ---------------- gather target rows ---------------------------------------
__global__ __launch_bounds__(256)
void gather_tgt(const float* __restrict__ NE, const int* __restrict__ TIDX,
                float* __restrict__ TGT)
{
    const int b   = blockIdx.x;
    const int tix = TIDX[b];
    const float* src = NE + ((size_t)b * NNODE + tix) * DIM;
    for (int k = threadIdx.x; k < DIM; k += 256)
        TGT[(size_t)b * DIM + k] = src[k];
}

// ---------------- attention: scores, softmax, C ----------------------------
__global__ __launch_bounds__(256)
void decoder_attn(const float* __restrict__ NE, const float* __restrict__ Sg,
                  float* __restrict__ Cg)
{
    extern __shared__ float lds[];
    float* E   = lds;            // 100*512 = 51200
    float* S   = E + 51200;      // 8*512   = 4096
    float* att = S + 4096;       // 8*104   = 832

    const int b    = blockIdx.x;
    const int tid  = threadIdx.x;
    const int lane = tid & 31;
    const int wid  = tid >> 5;

    const float4* src = (const float4*)(NE + (size_t)b * NNODE * DIM);
    for (int idx = tid; idx < NNODE * (DIM / 4); idx += 256)
        ((float4*)E)[idx] = src[idx];
    for (int i = tid; i < 8 * DIM; i += 256)
        S[i] = Sg[(size_t)b * (8 * DIM) + i];
    __syncthreads();

    // scores[h][n] = 0.125 * S[h].E[n]; two nodes per wave pass
    for (int p = wid; p < NNODE / 2; p += 8) {
        const int n0 = 2 * p, n1 = 2 * p + 1;
        float e0[16], e1[16];
        const float* er0 = E + n0 * DIM;
        const float* er1 = E + n1 * DIM;
#pragma unroll
        for (int t = 0; t < 16; ++t) {
            e0[t] = er0[lane + 32 * t];
            e1[t] = er1[lane + 32 * t];
        }
        for (int h = 0; h < 8; ++h) {
            const float* sr = S + h * DIM;
            float a0 = 0.f, a1 = 0.f;
#pragma unroll
            for (int t = 0; t < 16; ++t) {
                const float s = sr[lane + 32 * t];
                a0 += e0[t] * s;
                a1 += e1[t] * s;
            }
            a0 = wave_reduce_sum(a0);
            a1 = wave_reduce_sum(a1);
            if (lane == 0) {
                att[h * 104 + n0] = a0 * 0.125f;   // 1/sqrt(64)
                att[h * 104 + n1] = a1 * 0.125f;
            }
        }
    }
    __syncthreads();

    // per-head softmax over n (unmasked, as in reference MHA)
    if (wid < 8) {
        float* row = att + wid * 104;
        float m = -3.4e38f;
        for (int n = lane; n < NNODE; n += 32) m = fmaxf(m, row[n]);
        m = wave_reduce_max(m);
        float s = 0.f;
        for (int n = lane; n < NNODE; n += 32) {
            const float t = expf(row[n] - m);
            row[n] = t;
            s += t;
        }
        s = wave_reduce_sum(s);
        const float inv = 1.f / s;
        for (int n = lane; n < NNODE; n += 32) row[n] *= inv;
    }
    __syncthreads();

    // C[b][h*512+j] = sum_n attn[h][n] * E[n][j]  (overlays S's global slot)
    for (int j = tid; j < DIM; j += 256) {
        float acc[8] = {0, 0, 0, 0, 0, 0, 0, 0};
        for (int n = 0; n < NNODE; ++n) {
            const float e = E[n * DIM + j];
#pragma unroll
            for (int h = 0; h < 8; ++h) acc[h] += att[h * 104 + n] * e;
        }
#pragma unroll
        for (int h = 0; h < 8; ++h)
            Cg[(size_t)b * (8 * DIM) + h * DIM + j] = acc[h];
    }
}

// ---------------- logits + log_softmax -------------------------------------
__global__ __launch_bounds__(256)
void decoder_logits(const float* __restrict__ NE, const float* __restrict__ QK2,
                    const int* __restrict__ MASK, float* __restrict__ OUT)
{
    __shared__ float q2[DIM];
    __shared__ float lg[128];

    const int b    = blockIdx.x;
    const int tid  = threadIdx.x;
    const int lane = tid & 31;
    const int wid  = tid >> 5;

    for (int i = tid; i < DIM; i += 256) q2[i] = QK2[(size_t)b * DIM + i];
    __syncthreads();

    for (int n = wid; n < NNODE; n += 8) {
        const float* er = NE + ((size_t)b * NNODE + n) * DIM;
        float acc = 0.f;
#pragma unroll
        for (int t = 0; t < 16; ++t) {
            const int k = lane + 32 * t;
            acc += er[k] * q2[k];
        }
        acc = wave_reduce_sum(acc);
        if (lane == 0) {
            float l = 10.f * tanhf(acc);
            l -= (float)MASK[(size_t)b * NNODE + n] * 1e9f;
            lg[n] = l;
        }
    }
    __syncthreads();

    if (wid == 0) {
        float m = -3.4e38f;
        for (int n = lane; n < NNODE; n += 32) m = fmaxf(m, lg[n]);
        m = wave_reduce_max(m);
        float s = 0.f;
        for (int n = lane; n < NNODE; n += 32) s += expf(lg[n] - m);
        s = wave_reduce_sum(s);
        const float lz = m + logf(s);
        for (int n = lane; n < NNODE; n += 32)
            OUT[(size_t)b * NNODE + n] = lg[n] - lz;
    }
}

extern "C" void kernel_launch(void* const* d_in, const int* in_sizes, int n_in,
                              void* d_out, int out_size, void* d_ws, size_t ws_size,
                              hipStream_t stream)
{
    const float* NE   = (const float*)d_in[0];
    const float* GE   = (const float*)d_in[1];
    const int*   TIDX = (const int*)d_in[2];
    const int*   MASK = (const int*)d_in[3];
    const float* Wt   = (const float*)d_in[4];
    const float* Wg   = (const float*)d_in[5];
    const float* Wk1  = (const float*)d_in[6];
    const float* Wk2  = (const float*)d_in[7];
    const float* Wq   = (const float*)d_in[8];
    const float* Wv   = (const float*)d_in[9];
    const float* Wqm  = (const float*)d_in[10];
    const float* Wkm  = (const float*)d_in[11];
    const float* Wvm  = (const float*)d_in[12];
    const float* Wom  = (const float*)d_in[13];

    float* ws     = (float*)d_ws;
    float* wkcomb = ws;                 // 512x512 each
    float* wvT    = ws + 262144;
    float* a1T    = ws + 524288;
    float* a2T    = ws + 786432;
    float* ffold  = ws + 1048576;
    float* m1     = ws + 1310720;
    float* TGT    = ws + 1572864;       // [2048,512]; reused as MHA
    float* Q      = ws + 2621440;       // [2048,512]; reused as QK2
    float* S      = ws + 3670016;       // [2048,8,512]; reused as Cbuf
    float* MHA    = TGT;
    float* QK2    = Q;
    float* Cbuf   = S;                  // total 48.2 MB

    prep1<<<640, 256, 0, stream>>>(Wt, Wg, Wk1, Wq, Wv, Wqm, Wkm, Wvm, Wom,
                                   wkcomb, wvT, m1, a1T, a2T);
    prep2<<<128, 256, 0, stream>>>(m1, Wk2, ffold);

    gather_tgt<<<BATCH, 256, 0, stream>>>(NE, TIDX, TGT);

    // Q = TGT@A1^T + G@A2^T : M=2048,N=512,K=512
    gemm_wmma<<<dim3(128, 1), 256, 0, stream>>>(TGT, DIM, 0, a1T, DIM, 0,
                                                Q, DIM, 0, 8, DIM,
                                                GE, a2T, 1.0f);
    // S_h = Q[:,64h:+64] @ Wk_comb[64h-block] : per head M=2048,N=512,K=64
    gemm_wmma<<<dim3(128, 8), 256, 0, stream>>>(Q, DIM, 64,
                                                wkcomb, DIM, 64 * DIM,
                                                S, 8 * DIM, DIM,
                                                8, 64, nullptr, nullptr, 1.0f);

    const size_t ldsA = (size_t)(51200 + 4096 + 832) * sizeof(float); // 224.5 KB
    (void)hipFuncSetAttribute(reinterpret_cast<const void*>(decoder_attn),
                              hipFuncAttributeMaxDynamicSharedMemorySize,
                              (int)ldsA);
    decoder_attn<<<BATCH, 256, ldsA, stream>>>(NE, S, Cbuf);

    // MHA[:,64h:+64] = C_h @ WvT[:,64h-block] : per head M=2048,N=64,K=512
    gemm_wmma<<<dim3(16, 8), 256, 0, stream>>>(Cbuf, 8 * DIM, DIM,
                                               wvT, DIM, 64,
                                               MHA, DIM, 64,
                                               1, DIM, nullptr, nullptr, 1.0f);
    // QK2 = MHA @ Ffold * 1/sqrt(512) : M=2048,N=512,K=512
    gemm_wmma<<<dim3(128, 1), 256, 0, stream>>>(MHA, DIM, 0, ffold, DIM, 0,
                                                QK2, DIM, 0, 8, DIM,
                                                nullptr, nullptr,
                                                0.04419417382415922f);

    decoder_logits<<<BATCH, 256, 0, stream>>>(NE, QK2, MASK, (float*)d_out);
}